// CardClassifier_39221641347328
// MI455X (gfx1250) — compile-verified
//
#include <hip/hip_runtime.h>
#include <hip/hip_bf16.h>
#include <math.h>

typedef __attribute__((ext_vector_type(2))) float v2f;
typedef __attribute__((ext_vector_type(4))) float v4f;
typedef __attribute__((ext_vector_type(8))) float v8f;
typedef __attribute__((ext_vector_type(4))) int v4i;

// gfx1250 async global->LDS DMA path, compile-time guarded.
#if defined(__has_builtin)
#if __has_builtin(__builtin_amdgcn_global_load_async_to_lds_b128) && \
    __has_builtin(__builtin_amdgcn_s_wait_asynccnt)
#define HAVE_ASYNC_LDS 1
#endif
#endif
#ifndef HAVE_ASYNC_LDS
#define HAVE_ASYNC_LDS 0
#endif

// D = A(16x4) * B(4x16) + C, full fp32 WMMA (CDNA5)
__device__ __forceinline__ v8f wmma4(v2f a, v2f b, v8f c) {
  return __builtin_amdgcn_wmma_f32_16x16x4_f32(
      /*neg_a=*/false, a, /*neg_b=*/false, b,
      /*c_mod=*/(short)0, c, /*reuse_a=*/false, /*reuse_b=*/false);
}

// ---------------------------------------------------------------------------
// Fused conv3x3(SAME) + ReLU + 2x2 maxpool, implicit GEMM on WMMA fp32.
// One wave = one tile (batch b, pooled row hp, 16-wide w tile, 16 couts).
// K-order: k = pos*CINP + ci (pos = kh*3+kw outer, ci inner in steps of 4)
// so all boundary/index math hoists out of the inner loop. CIN==3 padded to
// CINP=4 with a compile-time tail mask. All padding via clamped (always
// legal) addresses * 0/1 mask — branchless, full EXEC for WMMA.
// ---------------------------------------------------------------------------
template <int CIN, int COUT, int HIN>
__global__ __launch_bounds__(128) void conv_relu_pool(
    const float* __restrict__ x, const float* __restrict__ wgt,
    const float* __restrict__ bias, float* __restrict__ out) {
  constexpr int CINP = (CIN % 4) ? ((CIN + 3) & ~3) : CIN;
  constexpr bool TAIL = (CIN % 4) != 0;
  constexpr int HOUT = HIN / 2;
  constexpr int WT = HIN / 16;
  constexpr int CT = COUT / 16;
  constexpr int TPB = HOUT * WT * CT;
  constexpr size_t CH = (size_t)HIN * HIN;

  const int lane = threadIdx.x & 31;
  const int wave = blockIdx.x * (blockDim.x >> 5) + (threadIdx.x >> 5);
  const int b = wave / TPB;
  int t = wave - b * TPB;
  const int hp = t / (WT * CT);
  t -= hp * (WT * CT);
  const int wt = t / CT;
  const int ct = t - wt * CT;

  const int col = lane & 15;           // pixel (A rows) == cout (B cols)
  const int khalf = (lane >> 4) << 1;  // 0 | 2 per ISA fp32 A/B layout
  const int h0 = hp * 2, w0 = wt * 16, co0 = ct * 16;

  const float bb = bias[co0 + col];
  v8f acc0, acc1;
#pragma unroll
  for (int i = 0; i < 8; ++i) { acc0[i] = bb; acc1[i] = bb; }

  const float* xb = x + (size_t)b * CIN * CH;
  const float* wr = wgt + (size_t)(co0 + col) * (CIN * 9);

#pragma unroll
  for (int pos = 0; pos < 9; ++pos) {
    const int kh = pos / 3, kw = pos - kh * 3;
    const int ih0 = h0 + kh - 1;               // in [-1, HIN-1]
    const int iw = w0 + col + kw - 1;          // in [-1, HIN]
    const int iwc = min(max(iw, 0), HIN - 1);  // clamped (legal) address
    const float mw = (iw >= 0 && iw < HIN) ? 1.f : 0.f;
    const float m0 = (ih0 >= 0) ? mw : 0.f;
    const float m1 = (ih0 + 1 < HIN) ? mw : 0.f;
    const float* xr0 = xb + (size_t)max(ih0, 0) * HIN + iwc;
    const float* xr1 = xb + (size_t)min(ih0 + 1, HIN - 1) * HIN + iwc;
    const float* wp = wr + pos;

    for (int c0 = 0; c0 < CINP; c0 += 4) {
      const int ci0 = c0 + khalf;
      const int ci1 = ci0 + 1;
      float mk0 = 1.f, mk1 = 1.f;
      int c0a = ci0, c1a = ci1;
      if constexpr (TAIL) {
        mk0 = (ci0 < CIN) ? 1.f : 0.f;
        mk1 = (ci1 < CIN) ? 1.f : 0.f;
        c0a = min(ci0, CIN - 1);
        c1a = min(ci1, CIN - 1);
      }
      v2f a0, a1, bv;
      a0[0] = xr0[(size_t)c0a * CH] * (m0 * mk0);
      a0[1] = xr0[(size_t)c1a * CH] * (m0 * mk1);
      a1[0] = xr1[(size_t)c0a * CH] * (m1 * mk0);
      a1[1] = xr1[(size_t)c1a * CH] * (m1 * mk1);
      bv[0] = wp[c0a * 9] * mk0;
      bv[1] = wp[c1a * 9] * mk1;
      acc0 = wmma4(a0, bv, acc0);  // conv row h0
      acc1 = wmma4(a1, bv, acc1);  // conv row h0+1
    }
  }

  // ReLU + 2x2 maxpool in-register; one b128 store per lane.
  // D layout: vgpr r holds pixel m=r (lanes 0-15) / m=r+8 (lanes 16-31).
  const int co = co0 + col;
  const int wp0 = wt * 8 + ((lane >> 4) << 2);
  float* op = out + ((size_t)(b * COUT + co) * HOUT + hp) * HOUT + wp0;
  v4f pv;
#pragma unroll
  for (int r = 0; r < 4; ++r) {
    const float p0 = fmaxf(fmaxf(acc0[2 * r], acc0[2 * r + 1]), 0.f);
    const float p1 = fmaxf(fmaxf(acc1[2 * r], acc1[2 * r + 1]), 0.f);
    pv[r] = fmaxf(p0, p1);
  }
  *(v4f*)op = pv;  // 16B aligned: wp0 % 4 == 0, row stride % 4 == 0
}

// ---------------------------------------------------------------------------
// Per-(b,f) sigmoid gate + input gating: one wave per (b,f); each lane owns
// 8 contiguous floats -> b128 loads/stores.
// ---------------------------------------------------------------------------
__global__ __launch_bounds__(256) void attention_gate(
    const float* __restrict__ feat, const float* __restrict__ Wa,
    const float* __restrict__ ba, float* __restrict__ zg) {
  const int lane = threadIdx.x & 31;
  const int wave = blockIdx.x * (blockDim.x >> 5) + (threadIdx.x >> 5);
  const int f = wave & 127;
  const float* fp = feat + (size_t)wave * 256 + lane * 8;
  const float* wp = Wa + (size_t)f * 256 + lane * 8;
  const v4f f0 = *(const v4f*)fp, f1 = *(const v4f*)(fp + 4);
  const v4f w0 = *(const v4f*)wp, w1 = *(const v4f*)(wp + 4);
  float s = 0.f;
#pragma unroll
  for (int i = 0; i < 4; ++i) s += f0[i] * w0[i] + f1[i] * w1[i];
#pragma unroll
  for (int off = 16; off > 0; off >>= 1) s += __shfl_xor(s, off, 32);
  const float a = 1.f / (1.f + expf(-(s + ba[f])));
  float* zp = zg + (size_t)wave * 256 + lane * 8;
  v4f z0, z1;
#pragma unroll
  for (int i = 0; i < 4; ++i) { z0[i] = f0[i] * a; z1[i] = f1[i] * a; }
  *(v4f*)zp = z0;
  *(v4f*)(zp + 4) = z1;
}

// ---------------------------------------------------------------------------
// Batched per-expert GEMM: C[b,f,n] = act(A[b,f,:] @ W[f,:,n] + bias[f,n]).
// One BLOCK (8 waves) per (expert f, 16-col n tile); wave w = m-tile w.
// The KKx16 W tile is staged once into LDS (async DMA to LDS on gfx1250,
// guarded fallback: b128 load + ds_store) and shared by all 8 waves — 8x
// less global W traffic. B operands then come from LDS (conflict-free:
// lane halves hit disjoint bank ranges).
// ---------------------------------------------------------------------------
template <int N, int KK, bool RELU>
__global__ __launch_bounds__(256) void expert_gemm(
    const float* __restrict__ A, const float* __restrict__ W,
    const float* __restrict__ bias, float* __restrict__ C) {
  constexpr int F = 128;
  constexpr int NT = N / 16;
  __shared__ __align__(16) float ldsW[KK * 16];

  const int tid = threadIdx.x;
  const int lane = tid & 31;
  const int mt = tid >> 5;  // wave index == m tile (0..7)
  const int f = blockIdx.x / NT;
  const int nt = blockIdx.x - f * NT;
  const int n0 = nt * 16;

  // ---- stage W[f, :, n0:n0+16] into LDS ----
  const float* Wf = W + (size_t)f * KK * N + n0;
  constexpr int QUADS = KK * 4;  // 4-float quads in the KKx16 tile
#pragma unroll
  for (int idx = tid; idx < QUADS; idx += 256) {
    const int k = idx >> 2, q = idx & 3;
    const float* src = Wf + (size_t)k * N + q * 4;  // 16B aligned
#if HAVE_ASYNC_LDS
    __builtin_amdgcn_global_load_async_to_lds_b128(
        (__attribute__((address_space(1))) v4i*)src,
        (__attribute__((address_space(3))) v4i*)&ldsW[idx * 4],
        /*offset=*/0, /*cpol=*/0);
#else
    *(v4f*)&ldsW[idx * 4] = *(const v4f*)src;
#endif
  }
#if HAVE_ASYNC_LDS
  __builtin_amdgcn_s_wait_asynccnt(0);
#endif
  __syncthreads();

  const int col = lane & 15;
  const int khalf = (lane >> 4) << 1;
  const int n = n0 + col;

  const float* Ap = A + ((size_t)(mt * 16 + col) * F + f) * KK + khalf;
  const float* Bl = &ldsW[khalf * 16 + col];
  const float bb = bias[f * N + n];
  v8f acc;
#pragma unroll
  for (int i = 0; i < 8; ++i) acc[i] = bb;

#pragma unroll 4
  for (int k0 = 0; k0 < KK; k0 += 4) {
    const v2f a = *(const v2f*)(Ap + k0);  // K = k0+khalf, k0+khalf+1
    v2f bv;
    bv[0] = Bl[k0 * 16];
    bv[1] = Bl[k0 * 16 + 16];
    acc = wmma4(a, bv, acc);
  }

  const int rbase = mt * 16 + ((lane >> 4) << 3);
#pragma unroll
  for (int r = 0; r < 8; ++r) {
    float v = acc[r];
    if (RELU) v = fmaxf(v, 0.f);
    C[((size_t)(rbase + r) * F + f) * N + n] = v;
  }
}

// ---------------------------------------------------------------------------
// Final dense: out[128,53] = Zr[128,4096] @ Fw[4096,53] + fb. N tail (53..63)
// via loop-invariant clamped pointer + 0/1 mask — branchless loads, full
// EXEC for WMMA; only the final stores are predicated.
// ---------------------------------------------------------------------------
__global__ __launch_bounds__(128) void final_dense(
    const float* __restrict__ Zr, const float* __restrict__ Fw,
    const float* __restrict__ Fb, float* __restrict__ out) {
  const int lane = threadIdx.x & 31;
  const int wave = blockIdx.x * (blockDim.x >> 5) + (threadIdx.x >> 5);
  const int mt = wave >> 2;
  const int nt = wave & 3;
  const int col = lane & 15;
  const int khalf = (lane >> 4) << 1;
  const int n = nt * 16 + col;
  const bool nok = (n < 53);
  const float msk = nok ? 1.f : 0.f;

  const float* Ap = Zr + (size_t)(mt * 16 + col) * 4096 + khalf;
  const float* Wp = Fw + (size_t)khalf * 53 + (nok ? n : 0);  // always legal
  const float bb = nok ? Fb[n] : 0.f;
  v8f acc;
#pragma unroll
  for (int i = 0; i < 8; ++i) acc[i] = bb;

#pragma unroll 4
  for (int k0 = 0; k0 < 4096; k0 += 4) {
    const v2f a = *(const v2f*)(Ap + k0);
    v2f bv;
    bv[0] = Wp[(size_t)k0 * 53] * msk;
    bv[1] = Wp[(size_t)(k0 + 1) * 53] * msk;
    acc = wmma4(a, bv, acc);
  }
  if (nok) {
    const int rbase = mt * 16 + ((lane >> 4) << 3);
#pragma unroll
    for (int r = 0; r < 8; ++r) out[(size_t)(rbase + r) * 53 + n] = acc[r];
  }
}

// ---------------------------------------------------------------------------
// Launch: liveness-based workspace reuse, 112 MiB total.
//   [0 .. 16M)      pool1 (64MiB), later recycled: zg/z1/z2/zr
//   [16M .. 24M)    pool2 (32MiB)
//   [24M .. 28M)    feat  (16MiB)
// ---------------------------------------------------------------------------
extern "C" void kernel_launch(void* const* d_in, const int* in_sizes, int n_in,
                              void* d_out, int out_size, void* d_ws,
                              size_t ws_size, hipStream_t stream) {
  (void)in_sizes; (void)n_in; (void)out_size; (void)ws_size;
  const float* x   = (const float*)d_in[0];
  const float* cw1 = (const float*)d_in[1];
  const float* cb1 = (const float*)d_in[2];
  const float* cw2 = (const float*)d_in[3];
  const float* cb2 = (const float*)d_in[4];
  const float* cw3 = (const float*)d_in[5];
  const float* cb3 = (const float*)d_in[6];
  const float* Wa  = (const float*)d_in[7];
  const float* ba  = (const float*)d_in[8];
  const float* ew1 = (const float*)d_in[9];
  const float* eb1 = (const float*)d_in[10];
  const float* ew2 = (const float*)d_in[11];
  const float* eb2 = (const float*)d_in[12];
  const float* ew3 = (const float*)d_in[13];
  const float* eb3 = (const float*)d_in[14];
  const float* fw  = (const float*)d_in[15];
  const float* fb  = (const float*)d_in[16];
  float* out = (float*)d_out;

  float* ws = (float*)d_ws;
  float* pool1 = ws;             // 128*32*64*64 = 16,777,216 floats
  float* pool2 = ws + 16777216;  // 128*64*32*32 =  8,388,608 floats
  float* feat  = ws + 25165824;  // 128*128*16*16 = 4,194,304 floats
  // pool1 region recycled once conv2 has consumed it:
  float* zg = ws;                //  4,194,304 floats
  float* z1 = ws + 4194304;      //  4,194,304 floats
  float* z2 = ws + 8388608;      //  2,097,152 floats
  float* zr = ws + 12582912;     //    524,288 floats

  const dim3 blk128(128), blk256(256);
  // conv: waves = B * HOUT * (W/16) * (COUT/16); 4 waves/block
  conv_relu_pool<3, 32, 128><<<32768, blk128, 0, stream>>>(x, cw1, cb1, pool1);
  conv_relu_pool<32, 64, 64><<<16384, blk128, 0, stream>>>(pool1, cw2, cb2, pool2);
  conv_relu_pool<64, 128, 32><<<8192, blk128, 0, stream>>>(pool2, cw3, cb3, feat);
  attention_gate<<<2048, blk256, 0, stream>>>(feat, Wa, ba, zg);
  // expert GEMMs: one block (8 waves) per (f, n-tile)
  expert_gemm<256, 256, true><<<128 * 16, blk256, 0, stream>>>(zg, ew1, eb1, z1);
  expert_gemm<128, 256, true><<<128 * 8, blk256, 0, stream>>>(z1, ew2, eb2, z2);
  // outer ReLU of the reference is folded into the last expert layer:
  expert_gemm<32, 128, true><<<128 * 2, blk256, 0, stream>>>(z2, ew3, eb3, zr);
  final_dense<<<8, blk128, 0, stream>>>(zr, fw, fb, out);
}